// LiftSelfAttention_53730040873215
// MI455X (gfx1250) — compile-verified
//
#include <hip/hip_runtime.h>

// Problem constants (from reference)
#define B_     4
#define CIN_   3
#define W_     28
#define P_     784      // W*W
#define MID_   16
#define HEADS_ 8
#define G_     4
#define COUT_  64
#define HALF_  8
#define NU_    110      // 2*(2W-1)
#define KLT_   49       // P_/16 key tiles

typedef float v2f __attribute__((ext_vector_type(2)));
typedef float v8f __attribute__((ext_vector_type(8)));

__device__ __forceinline__ v8f wmma_f32_16x16x4(v2f a, v2f b, v8f c) {
  // (neg_a, A, neg_b, B, c_mod, C, reuse_a, reuse_b) — same 8-arg pattern as f16/bf16
  return __builtin_amdgcn_wmma_f32_16x16x4_f32(false, a, false, b, (short)0, c, false, false);
}

// ---------------------------------------------------------------------------
// Kernel 1: positional-embedding MLP -> E_row / E_col tables (110 x 8 each)
// blockIdx.x == 0 -> row params, == 1 -> col params
// ---------------------------------------------------------------------------
__global__ __launch_bounds__(128)
void embed_kernel(const float* __restrict__ rw1, const float* __restrict__ rb1,
                  const float* __restrict__ rg,  const float* __restrict__ rbeta,
                  const float* __restrict__ rw2, const float* __restrict__ rb2,
                  const float* __restrict__ cw1, const float* __restrict__ cb1,
                  const float* __restrict__ cg,  const float* __restrict__ cbeta,
                  const float* __restrict__ cw2, const float* __restrict__ cb2,
                  float* __restrict__ e_row, float* __restrict__ e_col) {
  int t = threadIdx.x;
  if (t >= NU_) return;
  const bool col = (blockIdx.x == 1);
  const float* W1 = col ? cw1 : rw1;  const float* B1 = col ? cb1 : rb1;
  const float* GM = col ? cg  : rg;   const float* BT = col ? cbeta : rbeta;
  const float* W2 = col ? cw2 : rw2;  const float* B2 = col ? cb2 : rb2;
  float*       E  = col ? e_col : e_row;

  const int n = 2 * W_ - 1;  // 55
  float u = (t < n) ? (-1.f + 2.f * (float)t / (float)(n - 1))
                    : -(-1.f + 2.f * (float)(t - n) / (float)(n - 1));
  float h[16];
  float mu = 0.f;
  #pragma unroll
  for (int j = 0; j < 16; ++j) { h[j] = u * W1[j] + B1[j]; mu += h[j]; }
  mu *= (1.f / 16.f);
  float var = 0.f;
  #pragma unroll
  for (int j = 0; j < 16; ++j) { float d = h[j] - mu; var += d * d; }
  var *= (1.f / 16.f);
  float rstd = rsqrtf(var + 1e-5f);
  #pragma unroll
  for (int j = 0; j < 16; ++j) {
    float z = GM[j] * (h[j] - mu) * rstd + BT[j];
    h[j] = z / (1.f + __expf(-z));           // Swish
  }
  #pragma unroll
  for (int j2 = 0; j2 < HALF_; ++j2) {
    float s = B2[j2];
    #pragma unroll
    for (int j = 0; j < 16; ++j) s += h[j] * W2[j2 * 16 + j];
    E[t * HALF_ + j2] = s;
  }
}

// ---------------------------------------------------------------------------
// Kernel 2: 1x1-conv QKV projection, stored as [b][h][pixel][mid] (mid fastest)
// ---------------------------------------------------------------------------
__global__ __launch_bounds__(256)
void qkv_kernel(const float* __restrict__ x,
                const float* __restrict__ wq, const float* __restrict__ bq,
                const float* __restrict__ wk, const float* __restrict__ bk,
                const float* __restrict__ wv, const float* __restrict__ bv,
                float* __restrict__ qb, float* __restrict__ kb, float* __restrict__ vb) {
  int gid = blockIdx.x * blockDim.x + threadIdx.x;     // (b*P_+p)*128 + o
  if (gid >= B_ * P_ * (MID_ * HEADS_)) return;
  int o  = gid & 127;
  int bp = gid >> 7;
  int p  = bp % P_;
  int b  = bp / P_;
  float x0 = x[(b * CIN_ + 0) * P_ + p];
  float x1 = x[(b * CIN_ + 1) * P_ + p];
  float x2 = x[(b * CIN_ + 2) * P_ + p];
  float qv = bq[o] + wq[o * 3] * x0 + wq[o * 3 + 1] * x1 + wq[o * 3 + 2] * x2;
  float kv = bk[o] + wk[o * 3] * x0 + wk[o * 3 + 1] * x1 + wk[o * 3 + 2] * x2;
  float vv = bv[o] + wv[o * 3] * x0 + wv[o * 3 + 1] * x1 + wv[o * 3 + 2] * x2;
  int mid = o >> 3, h = o & 7;                          // o = mid*HEADS + h
  int idx = ((b * HEADS_ + h) * P_ + p) * MID_ + mid;
  qb[idx] = qv; kb[idx] = kv; vb[idx] = vv;
}

// ---------------------------------------------------------------------------
// Kernel 3: fused attention (flash style), one wave per (b,h,g, 16-query tile)
//   content tile:  4x v_wmma_f32_16x16x4_f32  (K = 16 mid channels)
//   pos term:      gather E_row[ridx], E_col[cidx], 8-dim dots in C/D layout
//   softmax:       online, row stats via shfl_xor over 16-lane N groups
//   P·V:           C->A relayout through LDS, 4x v_wmma_f32_16x16x4_f32
// ---------------------------------------------------------------------------
__global__ __launch_bounds__(32)
void attn_kernel(const float* __restrict__ qbuf, const float* __restrict__ kbuf,
                 const float* __restrict__ vbuf,
                 const float* __restrict__ e_row, const float* __restrict__ e_col,
                 const int* __restrict__ ridx, const int* __restrict__ cidx,
                 float* __restrict__ ov) {
  __shared__ float s_q[16][16];
  __shared__ float s_p[16][17];   // +1 pad to dodge bank conflicts

  int t = blockIdx.x;
  int itile = t % KLT_;  t /= KLT_;
  int g = t % G_;        t /= G_;
  int h = t % HEADS_;
  int b = t / HEADS_;

  int lane = threadIdx.x;
  int half = lane >> 4;        // 0: M rows 0-7 / K 0,1 ; 1: M rows 8-15 / K 2,3
  int l16  = lane & 15;
  int sel0 = half * 2;
  int sel1 = half * 2 + 1;

  const int qb0 = itile * 16;
  const float* qptr  = qbuf + ((size_t)(b * HEADS_ + h) * P_ + qb0) * MID_;
  const float* kbase = kbuf + (size_t)(b * HEADS_ + h) * P_ * MID_;
  const float* vbase = vbuf + (size_t)(b * HEADS_ + h) * P_ * MID_;

  // stage the 16x16 Q tile in LDS for the positional dot products
  for (int idx = lane; idx < 256; idx += 32) (&s_q[0][0])[idx] = qptr[idx];
  __syncthreads();

  // Q tile in A-matrix layout (M = l16, K chunks of 4)
  v2f aq[4];
  #pragma unroll
  for (int kk = 0; kk < 4; ++kk) {
    aq[kk].x = qptr[l16 * MID_ + 4 * kk + sel0];
    aq[kk].y = qptr[l16 * MID_ + 4 * kk + sel1];
  }

  float mrun[8], lrun[8];
  v8f acc = {};
  #pragma unroll
  for (int r = 0; r < 8; ++r) { mrun[r] = -__builtin_inff(); lrun[r] = 0.f; }

  const float inv_sqrt_c = 0.57735026918962576f;   // 1/sqrt(CIN)

  for (int kt = 0; kt < KLT_; ++kt) {
    const int klb = kt * 16;
    const float* kptr = kbase + (size_t)klb * MID_;

    // ---- content tile: S = Q(16x16) . K^T(16x16) via 4 chained WMMAs ----
    v8f s = {};
    #pragma unroll
    for (int kk = 0; kk < 4; ++kk) {
      v2f bk_;
      bk_.x = kptr[l16 * MID_ + 4 * kk + sel0];
      bk_.y = kptr[l16 * MID_ + 4 * kk + sel1];
      s = wmma_f32_16x16x4(aq[kk], bk_, s);
    }

    // ---- positional term, element-wise in C/D layout ----
    float sv[8];
    const int kj = klb + l16;
    #pragma unroll
    for (int r = 0; r < 8; ++r) {
      int M  = r + half * 8;
      int qi = qb0 + M;
      long pi = ((long)g * P_ + qi) * P_ + kj;          // flat (g,i,j,k,l)
      int ri = ridx[pi];
      int ci = cidx[pi];
      const float* er = e_row + (size_t)ri * HALF_;
      const float* ec = e_col + (size_t)ci * HALF_;
      float ps = 0.f;
      #pragma unroll
      for (int m = 0; m < HALF_; ++m)
        ps += s_q[M][m] * er[m] + s_q[M][HALF_ + m] * ec[m];
      sv[r] = (s[r] + ps) * inv_sqrt_c;
    }

    // ---- online softmax (rows span lanes 0-15 / 16-31 of each VGPR) ----
    float p_r[8];
    #pragma unroll
    for (int r = 0; r < 8; ++r) {
      float mx = sv[r];
      #pragma unroll
      for (int off = 8; off >= 1; off >>= 1)
        mx = fmaxf(mx, __shfl_xor(mx, off, 32));
      float mnew = fmaxf(mrun[r], mx);
      float corr = __expf(mrun[r] - mnew);
      float p    = __expf(sv[r] - mnew);
      float rs = p;
      #pragma unroll
      for (int off = 8; off >= 1; off >>= 1)
        rs += __shfl_xor(rs, off, 32);
      lrun[r] = lrun[r] * corr + rs;
      mrun[r] = mnew;
      acc[r]  = acc[r] * corr;
      p_r[r]  = p;
    }

    // ---- relayout P (C/D -> A) through LDS ----
    __syncthreads();
    #pragma unroll
    for (int r = 0; r < 8; ++r)
      s_p[r + half * 8][l16] = p_r[r];
    __syncthreads();

    // ---- acc += P(16x16) . V(16x16)  (N = value channel) ----
    #pragma unroll
    for (int kk = 0; kk < 4; ++kk) {
      v2f ap, bv_;
      ap.x  = s_p[l16][4 * kk + sel0];
      ap.y  = s_p[l16][4 * kk + sel1];
      bv_.x = vbase[(size_t)(klb + 4 * kk + sel0) * MID_ + l16];
      bv_.y = vbase[(size_t)(klb + 4 * kk + sel1) * MID_ + l16];
      acc = wmma_f32_16x16x4(ap, bv_, acc);
    }
  }

  // ---- normalize and store out_v as [b][h][g][pixel][mid] ----
  float* ovp = ov + (((size_t)(b * HEADS_ + h) * G_ + g) * P_) * MID_;
  #pragma unroll
  for (int r = 0; r < 8; ++r) {
    int qi = qb0 + r + half * 8;
    ovp[(size_t)qi * MID_ + l16] = acc[r] / lrun[r];
  }
}

// ---------------------------------------------------------------------------
// Kernel 4: final projection out[b,o,g,p] = b_out[o] + sum_{mid,h} w_out[o, mid*8+h] * ov
// ---------------------------------------------------------------------------
__global__ __launch_bounds__(256)
void proj_kernel(const float* __restrict__ ov, const float* __restrict__ w_out,
                 const float* __restrict__ b_out, float* __restrict__ out) {
  __shared__ float s_w[COUT_ * 128];
  int tid = threadIdx.x;
  for (int i = tid; i < COUT_ * 128; i += 256) s_w[i] = w_out[i];
  __syncthreads();

  int o   = tid & 63;
  int pos = blockIdx.x * 4 + (tid >> 6);     // (b,g,p) flat, 12544 total
  if (pos >= B_ * G_ * P_) return;
  int p = pos % P_;
  int g = (pos / P_) % G_;
  int b = pos / (P_ * G_);

  float s = b_out[o];
  #pragma unroll
  for (int h = 0; h < HEADS_; ++h) {
    const float* ovp = ov + (((size_t)(b * HEADS_ + h) * G_ + g) * P_ + p) * MID_;
    #pragma unroll
    for (int mid = 0; mid < MID_; ++mid)
      s += s_w[o * 128 + mid * HEADS_ + h] * ovp[mid];
  }
  out[((size_t)(b * COUT_ + o) * G_ + g) * P_ + p] = s;
}

// ---------------------------------------------------------------------------
extern "C" void kernel_launch(void* const* d_in, const int* in_sizes, int n_in,
                              void* d_out, int out_size, void* d_ws, size_t ws_size,
                              hipStream_t stream) {
  (void)in_sizes; (void)n_in; (void)out_size; (void)ws_size;

  const float* x     = (const float*)d_in[0];
  const float* wq    = (const float*)d_in[1];
  const float* bq    = (const float*)d_in[2];
  const float* wk    = (const float*)d_in[3];
  const float* bk    = (const float*)d_in[4];
  const float* wv    = (const float*)d_in[5];
  const float* bv    = (const float*)d_in[6];
  const float* w_out = (const float*)d_in[7];
  const float* b_out = (const float*)d_in[8];
  const float* rw1   = (const float*)d_in[9];
  const float* rb1   = (const float*)d_in[10];
  const float* rg    = (const float*)d_in[11];
  const float* rbe   = (const float*)d_in[12];
  const float* rw2   = (const float*)d_in[13];
  const float* rb2   = (const float*)d_in[14];
  const float* cw1   = (const float*)d_in[15];
  const float* cb1   = (const float*)d_in[16];
  const float* cg    = (const float*)d_in[17];
  const float* cbe   = (const float*)d_in[18];
  const float* cw2   = (const float*)d_in[19];
  const float* cb2   = (const float*)d_in[20];
  const int*   ridx  = (const int*)d_in[21];
  const int*   cidx  = (const int*)d_in[22];
  float*       out   = (float*)d_out;

  // workspace carve-up (floats)
  float* ws    = (float*)d_ws;
  float* e_row = ws;                              // 110*8      =     880
  float* e_col = e_row + NU_ * HALF_;             // 110*8      =     880
  float* qb    = e_col + NU_ * HALF_;             // 4*8*784*16 =  401408
  float* kb    = qb + B_ * HEADS_ * P_ * MID_;
  float* vb    = kb + B_ * HEADS_ * P_ * MID_;
  float* ov    = vb + B_ * HEADS_ * P_ * MID_;    // 4*8*4*784*16 = 1605632

  embed_kernel<<<2, 128, 0, stream>>>(rw1, rb1, rg, rbe, rw2, rb2,
                                      cw1, cb1, cg, cbe, cw2, cb2,
                                      e_row, e_col);

  int qkv_threads = B_ * P_ * (MID_ * HEADS_);    // 401408
  qkv_kernel<<<(qkv_threads + 255) / 256, 256, 0, stream>>>(
      x, wq, bq, wk, bk, wv, bv, qb, kb, vb);

  int attn_blocks = B_ * HEADS_ * G_ * KLT_;      // 6272 waves
  attn_kernel<<<attn_blocks, 32, 0, stream>>>(
      qb, kb, vb, e_row, e_col, ridx, cidx, ov);

  int proj_blocks = (B_ * G_ * P_) / 4;           // 3136
  proj_kernel<<<proj_blocks, 256, 0, stream>>>(ov, w_out, b_out, out);
}